// NearestCluster_29472065585602
// MI455X (gfx1250) — compile-verified
//
#include <hip/hip_runtime.h>

// NearestCluster on MI455X (gfx1250): per-batch NN via V_WMMA_F32_16X16X4_F32.
// Score(j,i) = |r_j|^2 - 2 q_i . r_j  (same argmin_j as squared distance).
// A row j   = (r_x, r_y, r_z, |r|^2)        [reference tile, 16x4]
// B col i   = (-2q_x, -2q_y, -2q_z, 1)      [query tile, 4x16]
// D[j][i]   = full score, 256 scores per WMMA, f32 exact.

typedef __attribute__((ext_vector_type(2))) float v2f;
typedef __attribute__((ext_vector_type(8))) float v8f;

#define L1V 4096
#define L2V 4096
#define NBATCH 8
#define JTILES (L1V / 16)

__global__ __launch_bounds__(256) void nearest_cluster_wmma_kernel(
    const float* __restrict__ coords1,   // [L1, N, 3] reference points
    const float* __restrict__ coords2,   // [L2, N, 3] query points
    float* __restrict__ out)             // [2 * L2 * N] (clusters, batch_idx)
{
    const int lane  = threadIdx.x & 31;
    const int wave  = (int)(blockIdx.x * (blockDim.x >> 5) + (threadIdx.x >> 5));
    const int n     = wave & (NBATCH - 1);   // batch index 0..7
    const int it    = wave >> 3;             // query tile 0..255

    const int  m  = lane & 15;
    const bool hi = lane >= 16;

    // ---------------- B fragment: query tile (columns) ----------------
    // VGPR0: lanes 0-15 -> K=0 (-2qx), lanes 16-31 -> K=2 (-2qz)
    // VGPR1: lanes 0-15 -> K=1 (-2qy), lanes 16-31 -> K=3 (1.0)
    const int    qi = it * 16 + m;
    const float* qp = coords2 + (qi * NBATCH + n) * 3;
    const float  qx = qp[0], qy = qp[1], qz = qp[2];
    v2f b;
    b.x = hi ? (-2.0f * qz) : (-2.0f * qx);
    b.y = hi ? 1.0f         : (-2.0f * qy);

    float best    = 3.402823466e+38f;
    int   bestIdx = 0;

    const float* rbase = coords1 + n * 3;

    for (int jt = 0; jt < JTILES; ++jt) {
        // ------------- A fragment: reference tile (rows) -------------
        // lanes 0-15: v0 = rx (K=0), v1 = ry (K=1)
        // lanes16-31: v0 = rz (K=2), v1 = |r|^2 (K=3)
        const int    rj = jt * 16 + m;
        const float* rp = rbase + rj * (NBATCH * 3);
        const float  rx = rp[0], ry = rp[1], rz = rp[2];
        const float  rs = rx * rx + ry * ry + rz * rz;
        v2f a;
        a.x = hi ? rz : rx;
        a.y = hi ? rs : ry;

        v8f c = {};
        // D = A(16x4) x B(4x16) + 0 : emits v_wmma_f32_16x16x4_f32
        c = __builtin_amdgcn_wmma_f32_16x16x4_f32(
                /*neg_a=*/false, a, /*neg_b=*/false, b,
                /*c_mod=*/(short)0, c, /*reuse_a=*/false, /*reuse_b=*/false);

        // Lane holds query column qi; VGPR k holds row jt*16 + k (+8 for hi half).
        const int jbase = jt * 16 + (hi ? 8 : 0);
#pragma unroll
        for (int k = 0; k < 8; ++k) {
            const float v   = c[k];
            const int   idx = jbase + k;
            const bool  lt  = v < best;          // strict < : first occurrence wins
            best    = lt ? v   : best;
            bestIdx = lt ? idx : bestIdx;
        }
    }

    // Combine halves: lane L has rows {0..7}+16t, lane L+16 has {8..15}+16t.
    const int osel   = (lane ^ 16) << 2;
    const int obestI = __builtin_amdgcn_ds_bpermute(osel, __float_as_int(best));
    const int oidx   = __builtin_amdgcn_ds_bpermute(osel, bestIdx);
    const float obf  = __int_as_float(obestI);
    const bool take  = (obf < best) || ((obf == best) && (oidx < bestIdx));
    best    = take ? obf  : best;
    bestIdx = take ? oidx : bestIdx;

    if (lane < 16) {
        // clusters[L2, N] flattened, then batch_idx[L2, N] flattened.
        out[qi * NBATCH + n]                 = (float)bestIdx;
        out[L2V * NBATCH + qi * NBATCH + n]  = (float)n;
    }
}

extern "C" void kernel_launch(void* const* d_in, const int* in_sizes, int n_in,
                              void* d_out, int out_size, void* d_ws, size_t ws_size,
                              hipStream_t stream) {
    const float* coords1 = (const float*)d_in[0];  // [4096, 8, 3]
    const float* coords2 = (const float*)d_in[1];  // [4096, 8, 3]
    float* out = (float*)d_out;                    // 2 * 4096 * 8 elements

    // 2048 waves (8 batches x 256 query tiles), 8 waves per 256-thread block.
    const int totalWaves = NBATCH * (L2V / 16);
    dim3 block(256);
    dim3 grid((totalWaves * 32) / 256);
    nearest_cluster_wmma_kernel<<<grid, block, 0, stream>>>(coords1, coords2, out);
}